// BalancedTopkModule_52003464020212
// MI455X (gfx1250) — compile-verified
//
#include <hip/hip_runtime.h>
#include <cstdint>

// BalancedTopk: per 128-wide bank row, keep top-16 of |x|+bias, zero the rest.
// One wave32 per row (4 elems/lane). Radix-1 MSB-down select via ballots with
// 64-bit packed popcounts (s_bcnt1_i32_b64) to unload the SALU pipe.
// Async global->LDS 8-deep prefetch ring per wave (gfx1250 ASYNCcnt path).

#define BANKW   128
#define NBANKS  32
#define HID     4096
#define TOPKN   16
#define TPW     16   // token-rows per wave
#define DEPTH   8    // async prefetch depth (rows in flight per wave)
#define WPB     8    // waves per block
#define BLOCKT  (WPB * 32)

// order-preserving float -> uint key (handles negative scores from bias)
__device__ __forceinline__ unsigned fkey(float s) {
    unsigned b = __float_as_uint(s);
    return b ^ ((unsigned)((int)b >> 31) | 0x80000000u);
}

template <int N>
__device__ __forceinline__ void wait_async() {
    asm volatile("s_wait_asynccnt %0" ::"n"(N) : "memory");
}

__device__ __forceinline__ void wait_ds0() {
    asm volatile("s_wait_dscnt 0x0" ::: "memory");
}

// async copy: 16B per lane from global -> LDS (512B per wave = one row)
__device__ __forceinline__ void async_row_load(const float* g, unsigned lds_byte) {
    asm volatile("global_load_async_to_lds_b128 %0, %1, off"
                 ::"v"(lds_byte), "v"(g)
                 : "memory");
}

__device__ __forceinline__ unsigned bal(bool p) {
    return __builtin_amdgcn_ballot_w32(p);
}

// two wave32 ballots -> one 64-bit popcount (s_bcnt1_i32_b64; the 64-bit
// construction is just SGPR-pair placement)
__device__ __forceinline__ unsigned pc2(unsigned lo, unsigned hi) {
    return (unsigned)__popcll((uint64_t)lo | ((uint64_t)hi << 32));
}

__global__ __launch_bounds__(BLOCKT) void topk_mask_kernel(
    const float* __restrict__ x, const float* __restrict__ bias,
    float* __restrict__ out, int tokens)
{
    __shared__ float4 smem[WPB * DEPTH * 32];  // 32 KB: 8 waves x 8 bufs x 512B

    const int lane = threadIdx.x & 31;
    const int wib  = threadIdx.x >> 5;
    const int gw   = blockIdx.x * WPB + wib;
    const int bank = gw & (NBANKS - 1);
    const int t0   = (gw >> 5) * TPW;
    if (t0 >= tokens) return;
    const int n = min(TPW, tokens - t0);

    // bias for this wave's bank: loaded once, lives in VGPRs
    const float4 b4 = *(const float4*)(bias + bank * BANKW + lane * 4);

    const size_t rowStride = (size_t)NBANKS * BANKW;  // 4096 floats per token
    const size_t base      = ((size_t)t0 * NBANKS + bank) * BANKW + (size_t)lane * 4;
    const float* gx   = x + base;
    float*       gout = out + base;

    float4* sm = &smem[(wib * DEPTH) * 32 + lane];
    const unsigned lds0 = (unsigned)(uintptr_t)sm;  // LDS byte offset of buffer 0

    auto process = [&](float4 xv, float* op) {
        // scores: |x| + bias  (abs folds into src modifier)
        const unsigned k0 = fkey(fabsf(xv.x) + b4.x);
        const unsigned k1 = fkey(fabsf(xv.y) + b4.y);
        const unsigned k2 = fkey(fabsf(xv.z) + b4.z);
        const unsigned k3 = fkey(fabsf(xv.w) + b4.w);

        // MSB-down radix-1 select: largest P with count(key >= P) >= 16.
        // Early-exit the moment some prefix threshold isolates exactly 16.
        unsigned P = 0u;
        for (unsigned b = 0x80000000u; b != 0u; b >>= 1) {
            const unsigned Q = P | b;
            const unsigned c = pc2(bal(k0 >= Q), bal(k1 >= Q)) +
                               pc2(bal(k2 >= Q), bal(k3 >= Q));
            if (c >= TOPKN) P = Q;
            if (c == TOPKN) break;
        }

        const unsigned cge = pc2(bal(k0 >= P), bal(k1 >= P)) +
                             pc2(bal(k2 >= P), bal(k3 >= P));

        float4 o;
        if (cge == TOPKN) {  // common case: threshold is exact
            o.x = (k0 >= P) ? xv.x : 0.0f;
            o.y = (k1 >= P) ? xv.y : 0.0f;
            o.z = (k2 >= P) ? xv.z : 0.0f;
            o.w = (k3 >= P) ? xv.w : 0.0f;
        } else {  // ties at P: keep strict-greater + first (16-g) equals by index
            const unsigned g = pc2(bal(k0 > P), bal(k1 > P)) +
                               pc2(bal(k2 > P), bal(k3 > P));
            const int quota = TOPKN - (int)g;
            const unsigned q0 = bal(k0 == P), q1 = bal(k1 == P);
            const unsigned q2 = bal(k2 == P), q3 = bal(k3 == P);
            const unsigned lt = (1u << lane) - 1u;
            const int before0 = __popc(q0 & lt) + __popc(q1 & lt) +
                                __popc(q2 & lt) + __popc(q3 & lt);
            const int before1 = before0 + (int)((q0 >> lane) & 1u);
            const int before2 = before1 + (int)((q1 >> lane) & 1u);
            const int before3 = before2 + (int)((q2 >> lane) & 1u);
            o.x = ((k0 > P) || ((k0 == P) && before0 < quota)) ? xv.x : 0.0f;
            o.y = ((k1 > P) || ((k1 == P) && before1 < quota)) ? xv.y : 0.0f;
            o.z = ((k2 > P) || ((k2 == P) && before2 < quota)) ? xv.z : 0.0f;
            o.w = ((k3 > P) || ((k3 == P) && before3 < quota)) ? xv.w : 0.0f;
        }
        *(float4*)op = o;  // global_store_b128
    };

    // ---- prime the async ring ----
    const int pre = min(DEPTH, n);
    for (int i = 0; i < pre; ++i)
        async_row_load(gx + (size_t)i * rowStride, lds0 + (unsigned)(i * 512));

    if (n == TPW) {
        // steady state: keep DEPTH rows in flight
        for (int i = 0; i < TPW - DEPTH; ++i) {
            wait_async<DEPTH - 1>();  // in-order completion => row i landed
            float4 xv = sm[(i % DEPTH) * 32];  // ds_load_b128
            process(xv, gout + (size_t)i * rowStride);
            wait_ds0();  // ds read retired before DMA reuses this buffer
            async_row_load(gx + (size_t)(i + DEPTH) * rowStride,
                           lds0 + (unsigned)((i % DEPTH) * 512));
        }
        // drain: everything already issued
        wait_async<0>();
        for (int j = 0; j < DEPTH; ++j) {
            const int i = TPW - DEPTH + j;
            float4 xv = sm[(i % DEPTH) * 32];
            process(xv, gout + (size_t)i * rowStride);
        }
    } else {
        // ragged tail (not hit for the harness shapes): conservative waits
        for (int i = 0; i < n; ++i) {
            wait_async<0>();
            float4 xv = sm[(i % DEPTH) * 32];
            process(xv, gout + (size_t)i * rowStride);
            wait_ds0();
            if (i + DEPTH < n)
                async_row_load(gx + (size_t)(i + DEPTH) * rowStride,
                               lds0 + (unsigned)((i % DEPTH) * 512));
        }
    }
}

// num_assigned_tokens is returned unchanged (eval mode): plain passthrough copy
__global__ void passthrough_copy(const float* __restrict__ src,
                                 float* __restrict__ dst, int n)
{
    int i = blockIdx.x * blockDim.x + threadIdx.x;
    if (i < n) dst[i] = src[i];
}

extern "C" void kernel_launch(void* const* d_in, const int* in_sizes, int n_in,
                              void* d_out, int out_size, void* d_ws, size_t ws_size,
                              hipStream_t stream) {
    (void)n_in; (void)d_ws; (void)ws_size;
    const float* x    = (const float*)d_in[0];
    const float* bias = (const float*)d_in[1];
    const float* nat  = (const float*)d_in[2];
    float*       out  = (float*)d_out;

    const int E      = in_sizes[0];     // 4*4096*4096
    const int tokens = E / HID;         // 16384
    const int natN   = in_sizes[2];     // 4096
    (void)out_size;

    const int tokenBlocks = (tokens + TPW - 1) / TPW;
    const int waves  = tokenBlocks * NBANKS;
    const int blocks = (waves + WPB - 1) / WPB;

    topk_mask_kernel<<<blocks, BLOCKT, 0, stream>>>(x, bias, out, tokens);
    passthrough_copy<<<(natN + 255) / 256, 256, 0, stream>>>(
        nat, out + (size_t)E, natN);
}